// BoundingBox_2834678415682
// MI455X (gfx1250) — compile-verified
//
#include <hip/hip_runtime.h>
#include <stdint.h>
#include <limits.h>

// Problem constants (reference: mask [128, 1, 512, 512] f32, thresh 0.5)
#define H 512
#define W 512
#define THRESH 0.5f
#define ROWS_PER_BLK 64
#define BLKS_PER_IMG (H / ROWS_PER_BLK)   // 8
#define THREADS 256                        // 8 wave32s
#define PASSES (ROWS_PER_BLK / 2)          // 2 rows per pass -> 32 passes
#define DEPTH 8                            // async pipeline depth (<= PASSES)

// ---------------------------------------------------------------------------
// CDNA5 async global->LDS stage: one GLOBAL_LOAD_ASYNC_TO_LDS_B128 per wave
// per call (32 lanes x 16B = 512B). GVS addressing: saddr(SGPR pair) +
// vaddr(32-bit byte offset). VDST VGPR carries the per-lane LDS byte address.
// The leading s_wait_dscnt 0 guarantees prior ds_loads from the buffer being
// recycled have completed before the async DMA may overwrite it.
// ---------------------------------------------------------------------------
__device__ __forceinline__ void async_stage(uint32_t lds_addr, uint32_t voff,
                                            const float* saddr) {
  asm volatile("s_wait_dscnt 0x0\n\t"
               "global_load_async_to_lds_b128 %0, %1, %2"
               :: "v"(lds_addr), "v"(voff), "s"(saddr)
               : "memory");
}

#define WAIT_ASYNC(n) asm volatile("s_wait_asynccnt " #n ::: "memory")

__global__ void bbox_init(int* __restrict__ ws, int n4) {
  int i = blockIdx.x * blockDim.x + threadIdx.x;
  if (i < n4) ws[i] = (i & 2) ? -1 : INT_MAX;   // [minR,minC,maxR,maxC] seeds
}

__global__ __launch_bounds__(THREADS)
void bbox_scan(const float* __restrict__ mask, int* __restrict__ ws) {
  __shared__ float4 stage[DEPTH][THREADS];   // 32 KB ring of stage buffers
  __shared__ int red[4];

  const int t   = threadIdx.x;
  const int img = blockIdx.y;
  const int r0  = blockIdx.x * ROWS_PER_BLK;
  const float* base = mask + (size_t)img * H * W;

  // Low 32 bits of a flat shared pointer == LDS byte address (aperture in hi32).
  const uint32_t ldsT    = (uint32_t)(uintptr_t)&stage[0][t];
  const uint32_t bufStep = (uint32_t)(THREADS * sizeof(float4));  // 4096 B
  // Thread t covers columns 4*(t%128).. in row r0 + (t/128) + 2*pass.
  const uint32_t voff0 =
      (uint32_t)(((r0 + (t >> 7)) * W + (t & 127) * 4) * sizeof(float));
  const uint32_t rowStep = 2u * W * sizeof(float);  // bytes per pass

  if (t == 0) { red[0] = INT_MAX; red[1] = INT_MAX; red[2] = -1; red[3] = -1; }

  // Prologue: fill all DEPTH buffers (DEPTH outstanding async ops per wave).
#pragma unroll
  for (int i = 0; i < DEPTH; ++i)
    async_stage(ldsT + (uint32_t)i * bufStep, voff0 + (uint32_t)i * rowStep,
                base);

  int minR = INT_MAX, minC = INT_MAX, maxR = -1, maxC = -1;
  const int cbase = (t & 127) * 4;

  auto process = [&](int p, const float4& v) {
    bool h0 = v.x >= THRESH, h1 = v.y >= THRESH;
    bool h2 = v.z >= THRESH, h3 = v.w >= THRESH;
    if (h0 | h1 | h2 | h3) {
      int row   = r0 + 2 * p + (t >> 7);
      int first = h0 ? 0 : (h1 ? 1 : (h2 ? 2 : 3));
      int last  = h3 ? 3 : (h2 ? 2 : (h1 ? 1 : 0));
      minR = min(minR, row);          maxR = max(maxR, row);
      minC = min(minC, cbase + first); maxC = max(maxC, cbase + last);
    }
  };

  // Steady state: async loads complete in order, so waiting until at most
  // DEPTH-1 remain outstanding guarantees buffer p&(DEPTH-1) is resident.
  for (int p = 0; p < PASSES - (DEPTH - 1); ++p) {
    WAIT_ASYNC(0x7);
    float4 v = stage[p & (DEPTH - 1)][t];
    process(p, v);
    if (p + DEPTH < PASSES)
      async_stage(ldsT + (uint32_t)(p & (DEPTH - 1)) * bufStep,
                  voff0 + (uint32_t)(p + DEPTH) * rowStep, base);
  }
  // Drain: decreasing literal waits for the final DEPTH-1 passes.
  {
    float4 v;
    WAIT_ASYNC(0x6); v = stage[(PASSES - 7) & 7][t]; process(PASSES - 7, v);
    WAIT_ASYNC(0x5); v = stage[(PASSES - 6) & 7][t]; process(PASSES - 6, v);
    WAIT_ASYNC(0x4); v = stage[(PASSES - 5) & 7][t]; process(PASSES - 5, v);
    WAIT_ASYNC(0x3); v = stage[(PASSES - 4) & 7][t]; process(PASSES - 4, v);
    WAIT_ASYNC(0x2); v = stage[(PASSES - 3) & 7][t]; process(PASSES - 3, v);
    WAIT_ASYNC(0x1); v = stage[(PASSES - 2) & 7][t]; process(PASSES - 2, v);
    WAIT_ASYNC(0x0); v = stage[(PASSES - 1) & 7][t]; process(PASSES - 1, v);
  }

  // wave32 shuffle reduction, then LDS atomics, then one global atomic set.
  for (int m = 16; m > 0; m >>= 1) {
    minR = min(minR, __shfl_xor(minR, m));
    minC = min(minC, __shfl_xor(minC, m));
    maxR = max(maxR, __shfl_xor(maxR, m));
    maxC = max(maxC, __shfl_xor(maxC, m));
  }
  __syncthreads();                      // red[] initialized before atomics
  if ((t & 31) == 0) {
    atomicMin(&red[0], minR); atomicMin(&red[1], minC);
    atomicMax(&red[2], maxR); atomicMax(&red[3], maxC);
  }
  __syncthreads();
  if (t == 0) {
    atomicMin(&ws[img * 4 + 0], red[0]);
    atomicMin(&ws[img * 4 + 1], red[1]);
    atomicMax(&ws[img * 4 + 2], red[2]);
    atomicMax(&ws[img * 4 + 3], red[3]);
  }
}

__global__ void bbox_final(const int* __restrict__ ws, int* __restrict__ out,
                           int n) {
  int i = blockIdx.x * blockDim.x + threadIdx.x;
  if (i < n) {
    int mnR = ws[4 * i + 0], mnC = ws[4 * i + 1];
    int mxR = ws[4 * i + 2], mxC = ws[4 * i + 3];
    bool any = (mxR >= 0);
    out[4 * i + 0] = any ? mnR : 0;        // y_min
    out[4 * i + 1] = any ? mnC : 0;        // x_min
    out[4 * i + 2] = any ? mxR + 1 : H;    // y_max
    out[4 * i + 3] = any ? mxC + 1 : W;    // x_max
  }
}

extern "C" void kernel_launch(void* const* d_in, const int* in_sizes, int n_in,
                              void* d_out, int out_size, void* d_ws,
                              size_t ws_size, hipStream_t stream) {
  (void)n_in; (void)out_size; (void)ws_size;
  const float* mask = (const float*)d_in[0];
  const int N = in_sizes[0] / (H * W);     // 128
  int* ws  = (int*)d_ws;                   // N*4 int32 scratch
  int* out = (int*)d_out;                  // int32 output [N,4]

  bbox_init<<<(N * 4 + 255) / 256, 256, 0, stream>>>(ws, N * 4);
  dim3 grid(BLKS_PER_IMG, N);
  bbox_scan<<<grid, THREADS, 0, stream>>>(mask, ws);
  bbox_final<<<(N + 127) / 128, 128, 0, stream>>>(ws, out, N);
}